// XSimGCL_49546742726724
// MI455X (gfx1250) — compile-verified
//
#include <hip/hip_runtime.h>
#include <hip/hip_bf16.h>
#include <math.h>

#define N_USER  100000
#define N_ITEM  50000
#define NN      (N_USER + N_ITEM)
#define EMB     64
#define LAYERS  3
#define NNZ     3000000
#define BATCH   2048
#define EPS_    0.1f
#define INV_TEMP 2.0f      // 1/0.5
#define CL_RATE 0.001f

typedef float v2f __attribute__((ext_vector_type(2)));
typedef float v8f __attribute__((ext_vector_type(8)));

__device__ __forceinline__ void atomAddF(float* p, float v) {
    // hardware global_atomic_add_f32 (L2-resident accumulation)
    unsafeAtomicAdd(p, v);
}

// ---------------- ego = concat(user_emb, item_emb); acc = ego ----------------
__global__ void k_init(const float* __restrict__ ue, const float* __restrict__ ie,
                       float* __restrict__ ego, float* __restrict__ acc) {
    size_t i = (size_t)blockIdx.x * blockDim.x + threadIdx.x;
    size_t total = (size_t)NN * EMB;
    if (i >= total) return;
    float v = (i < (size_t)N_USER * EMB) ? ue[i] : ie[i - (size_t)N_USER * EMB];
    ego[i] = v;
    acc[i] = v;
}

// ---------- newego[row] += val * ego[col]  (segment_sum of messages) ----------
// thread = one (edge, dim) pair; 32 lanes of a wave share one edge -> edge
// scalars are wave-uniform, dim accesses fully coalesced; atomics hit L2.
__global__ void k_spmm_scatter(const int* __restrict__ arow, const int* __restrict__ acol,
                               const float* __restrict__ aval,
                               const float* __restrict__ ego, float* __restrict__ out) {
    size_t gid = (size_t)blockIdx.x * blockDim.x + threadIdx.x;
    if (gid >= (size_t)NNZ * EMB) return;
    int e = (int)(gid >> 6);
    int d = (int)(gid & 63);
    int r = arow[e];
    int c = acol[e];
    float w = aval[e];
    atomAddF(&out[(size_t)r * EMB + d], w * ego[(size_t)c * EMB + d]);
}

// ---- ego = newego + l2norm(noise_k)*sign(newego)*EPS ; acc += ego ; aug? ----
__global__ void k_noise_acc(const float* __restrict__ noise_k, const float* __restrict__ newego,
                            float* __restrict__ ego, float* __restrict__ acc,
                            float* __restrict__ aug, int store_aug) {
    int r = blockIdx.x;          // node row
    int d = threadIdx.x;         // 0..63
    size_t idx = (size_t)r * EMB + d;
    float nv = noise_k[idx];
    __shared__ float red[EMB];
    red[d] = nv * nv;
    __syncthreads();
    for (int off = 32; off > 0; off >>= 1) {
        if (d < off) red[d] += red[d + off];
        __syncthreads();
    }
    float un = nv / fmaxf(sqrtf(red[0]), 1e-12f);
    float s  = newego[idx];
    float sg = (s > 0.f) ? 1.f : ((s < 0.f) ? -1.f : 0.f);
    float v  = s + un * sg * EPS_;
    ego[idx] = v;
    acc[idx] += v;
    if (store_aug) aug[idx] = v;
}

// -------- BPR: mean softplus(neg-pos) over batch (final = acc/4 folded) --------
__global__ void k_bpr(const float* __restrict__ acc, const int* __restrict__ users,
                      const int* __restrict__ pos, const int* __restrict__ neg,
                      float* __restrict__ out_sum) {
    int wave = (int)((blockIdx.x * blockDim.x + threadIdx.x) >> 5);
    int lane = threadIdx.x & 31;
    if (wave >= BATCH) return;
    const float* u = acc + (size_t)users[wave] * EMB;
    const float* p = acc + (size_t)(N_USER + pos[wave]) * EMB;
    const float* n = acc + (size_t)(N_USER + neg[wave]) * EMB;
    float u0 = u[lane], u1 = u[lane + 32];
    float dp = u0 * p[lane] + u1 * p[lane + 32];
    float dn = u0 * n[lane] + u1 * n[lane + 32];
    for (int off = 16; off > 0; off >>= 1) {
        dp += __shfl_xor(dp, off, 32);
        dn += __shfl_xor(dn, off, 32);
    }
    if (lane == 0) {
        float x  = (dn - dp) * 0.0625f;  // (1/4)*(1/4) from final = acc/4
        float sp = fmaxf(x, 0.f) + log1pf(expf(-fabsf(x)));  // stable softplus
        atomAddF(out_sum, sp);
    }
}

// ------- gather + unit-normalize the 4 InfoNCE views (scale of /4 cancels) ----
__global__ void k_gather_unit(const float* __restrict__ acc, const float* __restrict__ aug,
                              const int* __restrict__ users, const int* __restrict__ positems,
                              float* __restrict__ V) {
    int which = blockIdx.x / BATCH;  // 0: ue, 1: aug_ue, 2: ie, 3: aug_ie
    int b     = blockIdx.x % BATCH;
    int d     = threadIdx.x;
    const float* src = (which == 0 || which == 2) ? acc : aug;
    int rowi = (which < 2) ? users[b] : (N_USER + positems[b]);
    float v = src[(size_t)rowi * EMB + d];
    __shared__ float red[EMB];
    red[d] = v * v;
    __syncthreads();
    for (int off = 32; off > 0; off >>= 1) {
        if (d < off) red[d] += red[d + off];
        __syncthreads();
    }
    float nrm = fmaxf(sqrtf(red[0]), 1e-12f);
    V[((size_t)which * BATCH + b) * EMB + d] = v / nrm;
}

// ---- sim = V1 (2048x64) x V2^T via V_WMMA_F32_16X16X4_F32; fuse exp/rowsum ----
// One wave per 16x16 output tile; EXEC all-ones; sim never materialized.
__global__ void k_infonce_sim(const float* __restrict__ V1, const float* __restrict__ V2,
                              float* __restrict__ rowsum, float* __restrict__ diag) {
    const int TILES = BATCH / 16;                    // 128
    int wave = (int)((blockIdx.x * blockDim.x + threadIdx.x) >> 5);
    int lane = threadIdx.x & 31;
    int tm = wave / TILES;
    int tn = wave % TILES;
    int half = lane >> 4;        // 0: lanes 0-15, 1: lanes 16-31
    int mn   = lane & 15;
    // A frag lane holds A[m][2h],A[m][2h+1]; B frag lane holds B[2h][n],B[2h+1][n]
    // -> identical float2 addressing pattern into row-major V1/V2.
    const float* arow = V1 + ((size_t)(tm * 16 + mn)) * EMB + 2 * half;
    const float* brow = V2 + ((size_t)(tn * 16 + mn)) * EMB + 2 * half;
    v8f c = {};
#pragma unroll
    for (int kk = 0; kk < 16; ++kk) {
        v2f a, b;
        a.x = arow[kk * 4 + 0];
        a.y = arow[kk * 4 + 1];
        b.x = brow[kk * 4 + 0];
        b.y = brow[kk * 4 + 1];
        c = __builtin_amdgcn_wmma_f32_16x16x4_f32(
                /*neg_a=*/false, a, /*neg_b=*/false, b,
                /*c_mod=*/(short)0, c, /*reuse_a=*/false, /*reuse_b=*/false);
    }
    // C layout: VGPR r -> M = r + 8*half, N = lane&15
#pragma unroll
    for (int r = 0; r < 8; ++r) {
        int gm = tm * 16 + r + 8 * half;
        int gn = tn * 16 + mn;
        float s = expf(c[r] * INV_TEMP);
        if (gm == gn) diag[gm] = s;                  // unique writer per diagonal
        float t = s;                                 // reduce the 16 lanes sharing (r,half)
        t += __shfl_xor(t, 1, 32);
        t += __shfl_xor(t, 2, 32);
        t += __shfl_xor(t, 4, 32);
        t += __shfl_xor(t, 8, 32);
        if (mn == 0) atomAddF(&rowsum[gm], t);
    }
}

// ------------------------------- final reduction ------------------------------
__global__ void k_final(const float* __restrict__ bpr_sum,
                        const float* __restrict__ rsU, const float* __restrict__ dU,
                        const float* __restrict__ rsI, const float* __restrict__ dI,
                        float* __restrict__ out) {
    __shared__ float red[256];
    float a = 0.f;
    for (int r = threadIdx.x; r < BATCH; r += 256) {
        float pu = dU[r], su = rsU[r];
        float pi = dI[r], si = rsI[r];
        a += (logf(su - pu) - logf(pu)) + (logf(si - pi) - logf(pi));
    }
    red[threadIdx.x] = a;
    __syncthreads();
    for (int off = 128; off > 0; off >>= 1) {
        if (threadIdx.x < off) red[threadIdx.x] += red[threadIdx.x + off];
        __syncthreads();
    }
    if (threadIdx.x == 0)
        out[0] = bpr_sum[0] / (float)BATCH + CL_RATE * (red[0] / (float)BATCH);
}

extern "C" void kernel_launch(void* const* d_in, const int* in_sizes, int n_in,
                              void* d_out, int out_size, void* d_ws, size_t ws_size,
                              hipStream_t stream) {
    const float* user_emb  = (const float*)d_in[0];
    const float* item_emb  = (const float*)d_in[1];
    const int*   adj_row   = (const int*)d_in[2];
    const int*   adj_col   = (const int*)d_in[3];
    const float* adj_val   = (const float*)d_in[4];
    const float* noise     = (const float*)d_in[5];
    const int*   users     = (const int*)d_in[6];
    const int*   pos_items = (const int*)d_in[7];
    const int*   neg_items = (const int*)d_in[8];
    float* out = (float*)d_out;

    const size_t NE = (size_t)NN * EMB;  // 9.6M floats
    float* ws     = (float*)d_ws;
    float* ego    = ws;
    float* newego = ego + NE;
    float* acc    = newego + NE;
    float* aug    = acc + NE;
    float* V      = aug + NE;                    // 4 * BATCH * EMB
    float* rsU    = V + 4 * (size_t)BATCH * EMB;
    float* dU     = rsU + BATCH;
    float* rsI    = dU + BATCH;
    float* dI     = rsI + BATCH;
    float* bprsum = dI + BATCH;

    k_init<<<(unsigned)((NE + 255) / 256), 256, 0, stream>>>(user_emb, item_emb, ego, acc);

    for (int k = 0; k < LAYERS; ++k) {
        hipMemsetAsync(newego, 0, NE * sizeof(float), stream);
        k_spmm_scatter<<<(unsigned)(((size_t)NNZ * EMB + 255) / 256), 256, 0, stream>>>(
            adj_row, adj_col, adj_val, ego, newego);
        k_noise_acc<<<NN, EMB, 0, stream>>>(noise + (size_t)k * NE, newego,
                                            ego, acc, aug, (k == 0) ? 1 : 0);
    }

    hipMemsetAsync(rsU, 0, BATCH * sizeof(float), stream);
    hipMemsetAsync(rsI, 0, BATCH * sizeof(float), stream);
    hipMemsetAsync(bprsum, 0, sizeof(float), stream);

    k_bpr<<<BATCH / 8, 256, 0, stream>>>(acc, users, pos_items, neg_items, bprsum);
    k_gather_unit<<<4 * BATCH, EMB, 0, stream>>>(acc, aug, users, pos_items, V);

    const int TILES = BATCH / 16;                 // 128
    const int simBlocks = TILES * TILES / 8;      // 16384 waves / 8 per block
    k_infonce_sim<<<simBlocks, 256, 0, stream>>>(V, V + (size_t)BATCH * EMB, rsU, dU);
    k_infonce_sim<<<simBlocks, 256, 0, stream>>>(V + 2 * (size_t)BATCH * EMB,
                                                 V + 3 * (size_t)BATCH * EMB, rsI, dI);

    k_final<<<1, 256, 0, stream>>>(bprsum, rsU, dU, rsI, dI, out);
}